// SiameseLoss_70763881169378
// MI455X (gfx1250) — compile-verified
//
#include <hip/hip_runtime.h>

// ---------------------------------------------------------------------------
// Siamese loss, MI455X (gfx1250, wave32).
// HBM-bound (128 MB @ 23.3 TB/s ~ 5.6 us floor). Row reductions are done on
// the matrix unit: V_WMMA_F32_16X16X4_F32 with B == ones computes
// D[m][n] += sum_k A[m][k], i.e. per-row running sums at full f32 precision.
// ---------------------------------------------------------------------------

typedef float v2f __attribute__((ext_vector_type(2)));
typedef float v8f __attribute__((ext_vector_type(8)));

#define HALF       32768
#define DIM        512
#define COLS       32          // columns staged per LDS chunk
#define PITCH      36          // floats; multiple of 4 -> 16B-aligned rows, bank-conflict-free
#define TILE_ROWS  16          // pairs per wave
#define WPB        4           // waves per block

__global__ __launch_bounds__(128) void siamese_wmma_kernel(
    const float* __restrict__ x, const long long* __restrict__ label,
    float* __restrict__ partials)
{
    __shared__ float lds[WPB][2][TILE_ROWS * PITCH];

    const int wave = threadIdx.x >> 5;
    const int lane = threadIdx.x & 31;
    const int tile = blockIdx.x * WPB + wave;   // 0..2047
    const int p0   = tile * TILE_ROWS;          // first pair index of this tile

    float* t1 = lds[wave][0];
    float* t2 = lds[wave][1];

    v8f acc1 = {};                 // accumulates sum((x1-x2)^2) per row
    v8f acc2 = {};                 // accumulates sum((x1-x2*x2)^2) per row
    const v2f ones = {1.0f, 1.0f}; // B matrix: 4x16 of ones (layout-independent)

    // staging layout: 4 rows x (8 lanes * float4) per j-step, fully coalesced
    const int srow = lane >> 3;          // 0..3
    const int scol = (lane & 7) * 4;     // 0..28
    // A-matrix (16x4 f32) lane layout: lanes 0-15 -> K=0,1 ; lanes 16-31 -> K=2,3
    const int crow = lane & 15;
    const int koff = (lane >> 4) * 2;

    for (int cc = 0; cc < DIM / COLS; ++cc) {
        #pragma unroll
        for (int j = 0; j < 4; ++j) {
            const int r = j * 4 + srow;
            const size_t gcol = (size_t)cc * COLS + scol;
            const float4 g1 = *(const float4*)(x + (size_t)(p0 + r) * DIM + gcol);
            const float4 g2 = *(const float4*)(x + (size_t)(p0 + r + HALF) * DIM + gcol);
            *(float4*)(t1 + r * PITCH + scol) = g1;
            *(float4*)(t2 + r * PITCH + scol) = g2;
        }
        __syncthreads();

        #pragma unroll
        for (int kc = 0; kc < COLS / 4; ++kc) {
            const int col = kc * 4 + koff;
            const float2 a1 = *(const float2*)(t1 + crow * PITCH + col);
            const float2 a2 = *(const float2*)(t2 + crow * PITCH + col);
            const float dx = a1.x - a2.x;
            const float dy = a1.y - a2.y;
            const float ex = a1.x - a2.x * a2.x;
            const float ey = a1.y - a2.y * a2.y;
            v2f e1 = { dx * dx, dy * dy };
            v2f e2 = { ex * ex, ey * ey };
            // D = A x Ones + C : every output column accumulates the row sums
            acc1 = __builtin_amdgcn_wmma_f32_16x16x4_f32(
                false, e1, false, ones, (short)0, acc1, false, false);
            acc2 = __builtin_amdgcn_wmma_f32_16x16x4_f32(
                false, e2, false, ones, (short)0, acc2, false, false);
        }
        __syncthreads();
    }

    // C/D layout: lane l, VGPR r -> row (r + 8*(l>=16)), value replicated over N.
    const int base = (lane >> 4) * 8;
    float p = 0.0f;
    #pragma unroll
    for (int r = 0; r < 8; ++r) {
        const int m = p0 + base + r;
        const bool differ = label[m] != label[m + HALF];  // same-address loads coalesce
        const float s1 = acc1[r];
        const float s2 = acc2[r];
        const float same = fmaxf(1.0f - sqrtf(s2), 0.0f);
        p += differ ? s1 : same;
    }
    p += __shfl_xor(p, 16, 32);          // rows 0-7 + rows 8-15
    if (lane == 0) partials[tile] = p;
}

__global__ __launch_bounds__(256) void siamese_reduce_kernel(
    const float* __restrict__ partials, float* __restrict__ out)
{
    __shared__ float sm[256];
    float s = 0.0f;
    for (int i = threadIdx.x; i < 2048; i += 256) s += partials[i];
    sm[threadIdx.x] = s;
    __syncthreads();
    for (int off = 128; off > 0; off >>= 1) {
        if ((int)threadIdx.x < off) sm[threadIdx.x] += sm[threadIdx.x + off];
        __syncthreads();
    }
    if (threadIdx.x == 0) out[0] = sm[0] * (1.0f / 131072.0f);  // /(half*2*2)
}

extern "C" void kernel_launch(void* const* d_in, const int* in_sizes, int n_in,
                              void* d_out, int out_size, void* d_ws, size_t ws_size,
                              hipStream_t stream) {
    (void)in_sizes; (void)n_in; (void)out_size; (void)ws_size;
    const float*     x     = (const float*)d_in[0];
    const long long* label = (const long long*)d_in[1];   // int64 labels
    float* out      = (float*)d_out;
    float* partials = (float*)d_ws;                       // 2048 floats of scratch

    // 2048 tiles of 16 pairs, one tile per wave, 4 waves per block
    siamese_wmma_kernel<<<2048 / WPB, 32 * WPB, 0, stream>>>(x, label, partials);
    siamese_reduce_kernel<<<1, 256, 0, stream>>>(partials, out);
}